// EquivariantGraphNN_77163382440130
// MI455X (gfx1250) — compile-verified
//
#include <hip/hip_runtime.h>
#include <math.h>

typedef float v2f __attribute__((ext_vector_type(2)));
typedef float v8f __attribute__((ext_vector_type(8)));

// ---------------- degree / normalization ----------------

__global__ void gcn_deg_init(float* __restrict__ deg, int n) {
  int i = blockIdx.x * blockDim.x + threadIdx.x;
  if (i < n) deg[i] = 1.0f;  // self-loop contributes 1
}

__global__ void gcn_deg_accum(const int* __restrict__ dst, int E,
                              float* __restrict__ deg) {
  int e = blockIdx.x * blockDim.x + threadIdx.x;
  if (e < E) atomicAdd(&deg[dst[e]], 1.0f);
}

__global__ void gcn_deg_rsqrt(float* __restrict__ deg, int n) {
  int i = blockIdx.x * blockDim.x + threadIdx.x;
  if (i < n) {
    float d = deg[i];
    deg[i] = (d > 0.0f) ? rsqrtf(d) : 0.0f;
  }
}

// ---------------- fp32 WMMA GEMM:  H[M,F] = A[M,K] @ W[K,F] ----------------
// One wave computes one 16x16 tile of H using V_WMMA_F32_16X16X4_F32.
// A 16x4 tile: lanes 0-15 hold M=lane, VGPR{0,1}=K{0,1}; lanes 16-31 K{2,3}.
// B 4x16 tile: lanes hold N=lane%16, VGPR{0,1}=K{0,1} (lo half) / K{2,3} (hi).
// C/D: VGPR v, lane L -> M = v + 8*(L>=16), N = L%16.

__global__ __launch_bounds__(256) void gcn_gemm_wmma_f32(
    const float* __restrict__ A, const float* __restrict__ W,
    float* __restrict__ H, int M, int K, int F) {
  const int lane = threadIdx.x & 31;
  const int wave = threadIdx.x >> 5;
  const int ntiles = F >> 4;
  const int tile = blockIdx.x * 8 + wave;
  const int mtiles = M >> 4;           // M is a multiple of 16 (100000 = 6250*16)
  if (tile >= mtiles * ntiles) return; // whole-wave exit: EXEC stays all-1s for WMMA
  const int mt = tile / ntiles;
  const int nt = tile - mt * ntiles;

  const int half = lane >> 4;   // 0: K=0,1   1: K=2,3
  const int l16  = lane & 15;

  const float* arow = A + (size_t)(mt * 16 + l16) * K + 2 * half;
  const float* wcol = W + (size_t)(2 * half) * F + (nt * 16 + l16);

  v8f acc = {};
  for (int k0 = 0; k0 < K; k0 += 4) {
    v2f a = *(const v2f*)(arow + k0);          // A[row, k0+2h], A[row, k0+2h+1]
    v2f b;
    b.x = wcol[(size_t)k0 * F];                // W[k0+2h,   col]
    b.y = wcol[(size_t)k0 * F + F];            // W[k0+2h+1, col]
    acc = __builtin_amdgcn_wmma_f32_16x16x4_f32(
        /*neg_a=*/false, a, /*neg_b=*/false, b,
        /*c_mod=*/(short)0, acc, /*reuse_a=*/false, /*reuse_b=*/false);
  }

  float* hout = H + (size_t)(mt * 16 + 8 * half) * F + (nt * 16 + l16);
#pragma unroll
  for (int v = 0; v < 8; ++v) hout[(size_t)v * F] = acc[v];
}

// ---------------- aggregation ----------------
// out[i,f] = b[f] + h[i,f]*dinv[i]^2   (self-loop term; fully initializes out)
__global__ void gcn_agg_init(const float* __restrict__ h,
                             const float* __restrict__ bias,
                             const float* __restrict__ dinv,
                             float* __restrict__ out, int Nn, int F) {
  const int fc = F >> 2;
  int idx = blockIdx.x * blockDim.x + threadIdx.x;
  if (idx >= Nn * fc) return;
  int i = idx / fc, c = idx - i * fc;
  float d = dinv[i];
  float s = d * d;
  float4 hv = ((const float4*)h)[(size_t)i * fc + c];
  float4 bv = ((const float4*)bias)[c];
  float4 o;
  o.x = bv.x + hv.x * s;
  o.y = bv.y + hv.y * s;
  o.z = bv.z + hv.z * s;
  o.w = bv.w + hv.w * s;
  ((float4*)out)[(size_t)i * fc + c] = o;
}

// out[dst,f] += h[src,f] * dinv[src]*dinv[dst] over all edges.
// idx = e*fc + c so consecutive lanes stream one contiguous row of h (coalesced
// float4 gather from L2) and hit one contiguous destination row with atomics.
__global__ void gcn_agg_edges(const int* __restrict__ src,
                              const int* __restrict__ dst,
                              const float* __restrict__ h,
                              const float* __restrict__ dinv,
                              float* __restrict__ out, int E, int F) {
  const int fc = F >> 2;
  int idx = blockIdx.x * blockDim.x + threadIdx.x;
  if (idx >= E * fc) return;
  int e = idx / fc, c = idx - e * fc;
  int s = src[e], d = dst[e];
  float nrm = dinv[s] * dinv[d];
  float4 hv = ((const float4*)h)[(size_t)s * fc + c];
  float* op = out + (size_t)d * F + 4 * c;
  atomicAdd(op + 0, hv.x * nrm);
  atomicAdd(op + 1, hv.y * nrm);
  atomicAdd(op + 2, hv.z * nrm);
  atomicAdd(op + 3, hv.w * nrm);
}

__global__ void gcn_relu(float* __restrict__ x, int n) {
  int i = blockIdx.x * blockDim.x + threadIdx.x;
  if (i < n) x[i] = fmaxf(x[i], 0.0f);
}

// ---------------- launcher ----------------

static inline int cdiv(long long a, int b) { return (int)((a + b - 1) / b); }

extern "C" void kernel_launch(void* const* d_in, const int* in_sizes, int n_in,
                              void* d_out, int out_size, void* d_ws, size_t ws_size,
                              hipStream_t stream) {
  const float* x   = (const float*)d_in[0];
  const int*   ei  = (const int*)d_in[1];
  const float* W1  = (const float*)d_in[2];
  const float* b1  = (const float*)d_in[3];
  const float* W2  = (const float*)d_in[4];
  const float* b2  = (const float*)d_in[5];
  const float* W3  = (const float*)d_in[6];
  const float* b3  = (const float*)d_in[7];
  float* out = (float*)d_out;

  const int N = in_sizes[0] / 128;   // 100000
  const int E = in_sizes[1] / 2;     // 1600000
  const int* src = ei;               // edge_index[0]
  const int* dst = ei + E;           // edge_index[1]

  float* dinv = (float*)d_ws;                 // N floats
  float* bufH = dinv + N;                     // N*128 floats (GEMM output)
  float* bufA = bufH + (size_t)N * 128;       // N*128 floats (aggregated activ.)

  const int B = 256;

  // normalization: deg -> dinv
  gcn_deg_init<<<cdiv(N, B), B, 0, stream>>>(dinv, N);
  gcn_deg_accum<<<cdiv(E, B), B, 0, stream>>>(dst, E, dinv);
  gcn_deg_rsqrt<<<cdiv(N, B), B, 0, stream>>>(dinv, N);

  // ---- layer 1: 128 -> 64, ReLU ----
  {
    const int K = 128, F = 64;
    int tiles = (N >> 4) * (F >> 4);
    gcn_gemm_wmma_f32<<<cdiv(tiles, 8), 256, 0, stream>>>(x, W1, bufH, N, K, F);
    gcn_agg_init<<<cdiv((long long)N * (F >> 2), B), B, 0, stream>>>(bufH, b1, dinv, bufA, N, F);
    gcn_agg_edges<<<cdiv((long long)E * (F >> 2), B), B, 0, stream>>>(src, dst, bufH, dinv, bufA, E, F);
    gcn_relu<<<cdiv((long long)N * F, B), B, 0, stream>>>(bufA, N * F);
  }

  // ---- layer 2: 64 -> 128, ReLU ----
  {
    const int K = 64, F = 128;
    int tiles = (N >> 4) * (F >> 4);
    gcn_gemm_wmma_f32<<<cdiv(tiles, 8), 256, 0, stream>>>(bufA, W2, bufH, N, K, F);
    // bufA's layer-1 contents consumed by the GEMM above; safe to overwrite.
    gcn_agg_init<<<cdiv((long long)N * (F >> 2), B), B, 0, stream>>>(bufH, b2, dinv, bufA, N, F);
    gcn_agg_edges<<<cdiv((long long)E * (F >> 2), B), B, 0, stream>>>(src, dst, bufH, dinv, bufA, E, F);
    gcn_relu<<<cdiv((long long)N * F, B), B, 0, stream>>>(bufA, N * F);
  }

  // ---- layer 3: 128 -> 64, no ReLU, into d_out ----
  {
    const int K = 128, F = 64;
    int tiles = (N >> 4) * (F >> 4);
    gcn_gemm_wmma_f32<<<cdiv(tiles, 8), 256, 0, stream>>>(bufA, W3, bufH, N, K, F);
    gcn_agg_init<<<cdiv((long long)N * (F >> 2), B), B, 0, stream>>>(bufH, b3, dinv, out, N, F);
    gcn_agg_edges<<<cdiv((long long)E * (F >> 2), B), B, 0, stream>>>(src, dst, bufH, dinv, out, E, F);
  }
}